// ModelNew_58798102282503
// MI455X (gfx1250) — compile-verified
//
#include <hip/hip_runtime.h>
#include <hip/hip_bf16.h>

typedef __attribute__((ext_vector_type(2))) float v2f;
typedef __attribute__((ext_vector_type(8))) float v8f;

#define CIN_    256
#define COUT_   256
#define BATCH_  32
#define KDIM_   (CIN_ * 4)        // 1024
#define SPATIAL_ 16384.0f         // 128*128 output pixels
#define MULT_    2.0f

// ---------------------------------------------------------------------------
// Kernel 1: streaming reduction over x (the bandwidth-bound part, 134 MB).
// One wave per (b,c) image (64x64 = 16KB). Each lane does 32 float4 loads,
// then a 5-step xor-shuffle wave reduction (wave32).
// Produces A[b][c*4 + {0,1,2,3}] = {Sall, -R0, -C0, x00}.
// ---------------------------------------------------------------------------
__global__ void k_reduce_x(const float* __restrict__ x, float* __restrict__ A) {
  const int pair = blockIdx.x * 8 + (threadIdx.x >> 5);   // (b*256 + c)
  const int lane = threadIdx.x & 31;
  const float4* img = (const float4*)(x + (size_t)pair * 4096);

  float s = 0.f, r0 = 0.f, c0 = 0.f, x00 = 0.f;
#pragma unroll
  for (int it = 0; it < 32; ++it) {
    // global flat index g = it*128 + lane*4 ; h = g>>6 ; w0 = (lane*4)&63
    float4 v = img[it * 32 + lane];
    float vs = v.x + v.y + v.z + v.w;
    s += vs;
    if (it == 0 && lane < 16) r0 += vs;      // rows h==0 live in iter 0, lanes 0..15
    if ((lane & 15) == 0)     c0 += v.x;     // lanes 0,16 start at w==0 every iter
    if (it == 0 && lane == 0) x00 = v.x;     // x[b,c,0,0]
  }
#pragma unroll
  for (int off = 16; off > 0; off >>= 1) {
    s   += __shfl_xor(s,   off, 32);
    r0  += __shfl_xor(r0,  off, 32);
    c0  += __shfl_xor(c0,  off, 32);
    x00 += __shfl_xor(x00, off, 32);
  }
  if (lane == 0) {
    const int b = pair >> 8, c = pair & 255;
    float* dst = A + (size_t)b * KDIM_ + (size_t)c * 4;
    dst[0] = s; dst[1] = -r0; dst[2] = -c0; dst[3] = x00;
  }
}

// ---------------------------------------------------------------------------
// Kernel 2: weight prep (2.4 MB read, trivial). One thread per (c,o) pair.
// weight layout [c][o][kh][kw] -> Wp[(c*4+j)][o] row-major (K x N for the GEMM):
//   j=0: sum over all 9 taps, j=1: sum over kh==0 row, j=2: sum over kw==0 col,
//   j=3: w[0][0].
// ---------------------------------------------------------------------------
__global__ void k_prep_w(const float* __restrict__ w, float* __restrict__ Wp) {
  const int t = blockIdx.x * blockDim.x + threadIdx.x;   // 0..65535 == c*256+o
  const int c = t >> 8, o = t & 255;
  const float* p = w + (size_t)t * 9;
  float w0 = p[0], w1 = p[1], w2 = p[2], w3 = p[3], w4 = p[4],
        w5 = p[5], w6 = p[6], w7 = p[7], w8 = p[8];
  const float wsum = w0 + w1 + w2 + w3 + w4 + w5 + w6 + w7 + w8;
  const float wk0  = w0 + w1 + w2;   // kh == 0
  const float wl0  = w0 + w3 + w6;   // kw == 0
  Wp[(size_t)(c * 4 + 0) * COUT_ + o] = wsum;
  Wp[(size_t)(c * 4 + 1) * COUT_ + o] = wk0;
  Wp[(size_t)(c * 4 + 2) * COUT_ + o] = wl0;
  Wp[(size_t)(c * 4 + 3) * COUT_ + o] = w0;
}

// ---------------------------------------------------------------------------
// Kernel 3: S = A[32x1024] @ Wp[1024x256] via V_WMMA_F32_16X16X4_F32,
// fused epilogue out = 2*(S/16384 + bias). One wave per 16x16 tile,
// grid = 2 (M tiles) x 16 (N tiles) = 32 waves, 256 WMMAs each.
// A operand: lane L holds A[M = L&15][K0 + {0,1}], K0 = 2*(L>>4).
// B operand: lane L holds B[K0 + {0,1}][N = L&15].
// D: VGPR r holds D[r + 8*(L>>4)][L&15].
// ---------------------------------------------------------------------------
__global__ void k_gemm_wmma(const float* __restrict__ A, const float* __restrict__ Wp,
                            const float* __restrict__ bias, float* __restrict__ out) {
  const int lane  = threadIdx.x;          // 0..31, full wave, EXEC all-ones
  const int tileM = blockIdx.x & 1;       // 0..1
  const int tileN = blockIdx.x >> 1;      // 0..15
  const int mn    = lane & 15;
  const int khalf = (lane >> 4) * 2;

  const float* arow = A  + (size_t)(tileM * 16 + mn) * KDIM_ + khalf;
  const float* bcol = Wp + (size_t)khalf * COUT_ + (size_t)(tileN * 16 + mn);

  v8f acc = {};
#pragma unroll 4
  for (int k = 0; k < KDIM_; k += 4) {
    v2f a, b;
    a.x = arow[k];                         // contiguous pair -> b64 load
    a.y = arow[k + 1];
    b.x = bcol[(size_t)k * COUT_];         // stride-N column loads
    b.y = bcol[(size_t)(k + 1) * COUT_];
    acc = __builtin_amdgcn_wmma_f32_16x16x4_f32(
        /*neg_a=*/false, a, /*neg_b=*/false, b,
        /*c_mod=*/(short)0, acc, /*reuse_a=*/false, /*reuse_b=*/false);
  }

  const int o  = tileN * 16 + mn;
  const float bv = bias[o];
#pragma unroll
  for (int r = 0; r < 8; ++r) {
    const int m = r + 8 * (lane >> 4);
    const int b = tileM * 16 + m;
    out[(size_t)b * COUT_ + o] = MULT_ * (acc[r] / SPATIAL_ + bv);
  }
}

// ---------------------------------------------------------------------------
extern "C" void kernel_launch(void* const* d_in, const int* in_sizes, int n_in,
                              void* d_out, int out_size, void* d_ws, size_t ws_size,
                              hipStream_t stream) {
  const float* x    = (const float*)d_in[0];   // [32,256,64,64]
  const float* w    = (const float*)d_in[1];   // [256,256,3,3]
  const float* bias = (const float*)d_in[2];   // [256]
  float* out = (float*)d_out;                  // [32,256,1,1] flat

  float* A  = (float*)d_ws;                    // 32*1024 f32 = 128 KB
  float* Wp = A + (size_t)BATCH_ * KDIM_;      // 1024*256 f32 = 1 MB

  // Stage 1: reductions over x. 8192 (b,c) pairs, 8 waves per 256-thread block.
  k_reduce_x<<<(BATCH_ * CIN_) / 8, 256, 0, stream>>>(x, A);
  // Stage 2: weight prep. 65536 threads.
  k_prep_w<<<(CIN_ * COUT_) / 256, 256, 0, stream>>>(w, Wp);
  // Stage 3: WMMA GEMM + epilogue. 32 tiles, one wave each.
  k_gemm_wmma<<<32, 32, 0, stream>>>(A, Wp, bias, out);
}